// X_Transformer_22625887715608
// MI455X (gfx1250) — compile-verified
//
#include <hip/hip_runtime.h>
#include <hip/hip_bf16.h>
#include <math.h>

// ---------------------------------------------------------------------------
// Problem constants (match reference setup_inputs)
// ---------------------------------------------------------------------------
#define BATCH   2
#define SEQ     1024
#define DMODEL  512
#define NHEADS  8
#define HDIM    64
#define FFDIM   2048
#define NLAYERS 4
#define VOCAB   32000
#define ROWS    (BATCH * SEQ)          // 2048
#define QKV_LD  (3 * DMODEL)           // 1536
#define ATT_SCALE 0.125f               // 1/sqrt(64)
#define AW      8                      // waves per attention block
#define GN      128                    // GEMM block N-tile (floats)

typedef __attribute__((ext_vector_type(2))) float v2f;
typedef __attribute__((ext_vector_type(8))) float v8f;

static __device__ __forceinline__ v8f wmma_f32(v2f a, v2f b, v8f c) {
  // V_WMMA_F32_16X16X4_F32 : D = A(16x4) * B(4x16) + C(16x16)
  return __builtin_amdgcn_wmma_f32_16x16x4_f32(false, a, false, b, (short)0, c,
                                               false, false);
}

static __device__ __forceinline__ int permrow(int i, int permuted) {
  // h2[b, r*16 + n] = h[b, n*64 + r]  => perm(i) = (i%16)*64 + i/16
  return permuted ? (((i & 15) << 6) + (i >> 4)) : i;
}

// CDNA5 async global->LDS copy (16 bytes per lane), tracked by ASYNCcnt.
// vdst = 32-bit LDS byte address (flat LDS addresses keep offset in [31:0]).
static __device__ __forceinline__ void async_copy_b128(unsigned int lds_off,
                                                       const void* gaddr) {
  asm volatile("global_load_async_to_lds_b128 %0, %1, off"
               :: "v"(lds_off), "v"(gaddr)
               : "memory");
}
static __device__ __forceinline__ void wait_async_le2() {
  asm volatile("s_wait_asynccnt 0x2" ::: "memory");
}
static __device__ __forceinline__ void wait_async_le0() {
  asm volatile("s_wait_asynccnt 0x0" ::: "memory");
}

// ---------------------------------------------------------------------------
// Embedding: x[row] = embed_w[tok[row]] * sqrt(D) + pos_embed[row % SEQ]
// ---------------------------------------------------------------------------
__global__ void embed_kernel(const int* __restrict__ tok,
                             const float* __restrict__ ew,
                             const float* __restrict__ pe,
                             float* __restrict__ x) {
  const int row = blockIdx.x;
  const int tid = threadIdx.x;                 // 128 threads
  const int t = tok[row];
  const int s = row & (SEQ - 1);
  const float* e = ew + (size_t)t * DMODEL;
  const float* p = pe + (size_t)s * DMODEL;
  float* xr = x + (size_t)row * DMODEL;
  const float scale = 22.627416997969522f;     // sqrt(512)
#pragma unroll
  for (int k = 0; k < DMODEL / 128; ++k) {
    int c = tid + k * 128;
    xr[c] = e[c] * scale + p[c];
  }
}

__global__ void fill_zero(float* __restrict__ p, int n) {
  int i = blockIdx.x * blockDim.x + threadIdx.x;
  if (i < n) p[i] = 0.0f;
}

// ---------------------------------------------------------------------------
// fp32 WMMA GEMM with async double-buffered LDS staging of B.
// C[M,N] = A[M,K] @ B[K,N] (+bias) (+relu)
// Block = 256 threads (8 waves). Each wave computes a 32(M) x 128(N) strip;
// the 8 waves stack along M (256 M rows per block), all sharing each
// 16(K) x 128(N) B panel staged in LDS via GLOBAL_LOAD_ASYNC_TO_LDS_B128
// (ASYNCcnt), double-buffered. Each B fragment DS read feeds two WMMAs and
// each A fragment feeds eight, so per k-step: 2 global b64 + 8 DS reads
// drive 16 WMMAs. __launch_bounds__(256,1) gives the 16 v8f accumulators a
// full register budget (no scratch spills). M%256==0, N%128==0, K%16==0.
// ---------------------------------------------------------------------------
__global__ __launch_bounds__(256, 1)
void gemm_wmma(const float* __restrict__ A, const float* __restrict__ B,
               float* __restrict__ C, const float* __restrict__ bias,
               int K, int lda, int ldb, int ldc, int relu) {
  __shared__ float sB[2][16 * GN];             // two 8 KB B panels

  const int tid  = threadIdx.x;                // 0..255
  const int lane = tid & 31;
  const int wid  = tid >> 5;                   // 0..7 (wave id -> M sub-tile)
  const int half = lane >> 4;                  // 0/1
  const int mr   = lane & 15;
  const int n0 = blockIdx.x * GN;
  const int m0 = blockIdx.y * 256 + wid * 32;

  const unsigned int ldsBase0 = (unsigned int)(uintptr_t)(&sB[0][0]);
  const unsigned int ldsBase1 = (unsigned int)(uintptr_t)(&sB[1][0]);

  // Two float4 staging slots per thread (512 slots = 16x128 panel).
  const int q0 = tid, q1 = tid + 256;
  const int r0 = q0 >> 5, c0 = (q0 & 31) << 2; // panel row, col (floats)
  const int r1 = q1 >> 5, c1 = (q1 & 31) << 2;

  const float* arowL = A + (size_t)(m0 + mr) * lda + 2 * half;
  const float* arowH = arowL + (size_t)16 * lda;
  const v8f zero = {0.f, 0.f, 0.f, 0.f, 0.f, 0.f, 0.f, 0.f};
  v8f accL[8] = {zero, zero, zero, zero, zero, zero, zero, zero};
  v8f accH[8] = {zero, zero, zero, zero, zero, zero, zero, zero};

  const int nPanels = K >> 4;

  // Prime the pipeline: panel 0 -> buffer 0
  async_copy_b128(ldsBase0 + (unsigned)(q0 << 4), B + (size_t)r0 * ldb + n0 + c0);
  async_copy_b128(ldsBase0 + (unsigned)(q1 << 4), B + (size_t)r1 * ldb + n0 + c1);

  for (int p = 0; p < nPanels; ++p) {
    const int cur = p & 1;
    if (p + 1 < nPanels) {
      const int k0n = (p + 1) << 4;
      const unsigned int nb = (cur ? ldsBase0 : ldsBase1);
      async_copy_b128(nb + (unsigned)(q0 << 4),
                      B + (size_t)(k0n + r0) * ldb + n0 + c0);
      async_copy_b128(nb + (unsigned)(q1 << 4),
                      B + (size_t)(k0n + r1) * ldb + n0 + c1);
      wait_async_le2();                        // current panel complete
    } else {
      wait_async_le0();
    }
    __syncthreads();                           // publish LDS across waves

    const float* bufc = &sB[cur][0];
    const int kbase = p << 4;
#pragma unroll
    for (int kk = 0; kk < 16; kk += 4) {
      v2f aL = *reinterpret_cast<const v2f*>(arowL + kbase + kk);
      v2f aH = *reinterpret_cast<const v2f*>(arowH + kbase + kk);
      const int rb = (kk + 2 * half) * GN + mr;
#pragma unroll
      for (int nt = 0; nt < 8; ++nt) {
        v2f b;
        b[0] = bufc[rb + nt * 16];
        b[1] = bufc[rb + GN + nt * 16];
        accL[nt] = wmma_f32(aL, b, accL[nt]);
        accH[nt] = wmma_f32(aH, b, accH[nt]);
      }
    }
    __syncthreads();                           // done reading this buffer
  }

#pragma unroll
  for (int nt = 0; nt < 8; ++nt) {
    int col = n0 + nt * 16 + mr;
    float bv = bias ? bias[col] : 0.0f;
#pragma unroll
    for (int r = 0; r < 8; ++r) {
      int rowL = m0 + r + 8 * half;
      int rowH = rowL + 16;
      float vL = accL[nt][r] + bv;
      float vH = accH[nt][r] + bv;
      if (relu) { vL = fmaxf(vL, 0.0f); vH = fmaxf(vH, 0.0f); }
      C[(size_t)rowL * ldc + col] = vL;
      C[(size_t)rowH * ldc + col] = vH;
    }
  }
}

// ---------------------------------------------------------------------------
// Online-softmax update + P@V accumulation for one mask variant.
// pbuf is this wave's private 16x16 LDS tile (DS ops are in-order per wave,
// so no barrier is needed between the store and the A-layout reload).
// ---------------------------------------------------------------------------
static __device__ __forceinline__ void softmax_pv(
    const v8f& s, int maskBlk, int qt, int kt, int mr, int half,
    float (&mrow)[8], float (&lrow)[8], v8f (&acc)[4],
    float* __restrict__ pbuf, const float* __restrict__ vb) {
  const int j = kt * 16 + mr;                   // logical key index (column)
  float p[8];
#pragma unroll
  for (int r = 0; r < 8; ++r) {
    const int i = qt * 16 + r + 8 * half;       // logical query index (row)
    const bool msk = maskBlk ? ((j & 15) > (i & 15)) : ((j & 63) > (i & 63));
    float v = msk ? -INFINITY : s[r] * ATT_SCALE;
    float rmax = v;
#pragma unroll
    for (int o = 1; o < 16; o <<= 1) rmax = fmaxf(rmax, __shfl_xor(rmax, o, 32));
    const float mnew = fmaxf(mrow[r], rmax);
    const float corr = (mrow[r] == -INFINITY) ? 0.0f : __expf(mrow[r] - mnew);
    const float pe = (v == -INFINITY) ? 0.0f : __expf(v - mnew);
    float rsum = pe;
#pragma unroll
    for (int o = 1; o < 16; o <<= 1) rsum += __shfl_xor(rsum, o, 32);
    lrow[r] = lrow[r] * corr + rsum;
    mrow[r] = mnew;
    p[r] = pe;
#pragma unroll
    for (int nt = 0; nt < 4; ++nt) acc[nt][r] *= corr;
  }
  // P: C layout -> LDS -> A layout
#pragma unroll
  for (int r = 0; r < 8; ++r) pbuf[(r + 8 * half) * 16 + mr] = p[r];
#pragma unroll
  for (int ks = 0; ks < 16; ks += 4) {
    v2f ap = *reinterpret_cast<const v2f*>(&pbuf[mr * 16 + ks + 2 * half]);
    const float* v0 = vb + (ks + 2 * half) * 64 + mr;
    const float* v1 = v0 + 64;
#pragma unroll
    for (int nt = 0; nt < 4; ++nt) {
      v2f bv;
      bv[0] = v0[nt * 16];
      bv[1] = v1[nt * 16];
      acc[nt] = wmma_f32(ap, bv, acc[nt]);
    }
  }
}

// ---------------------------------------------------------------------------
// Fused flash-attention. Block = 8 waves, one 16-query tile per wave; all
// waves share K/V panels async-staged into LDS (double-buffered, ASYNCcnt).
// dual==1 (permuted==0): one Q@K^T feeds BOTH row-mask and blk-mask softmax
//   states (scores are identical), covering attentions #1 and #3.
// dual==0 (permuted==1): blk-mask on block-transposed rows (attention #2);
//   the permutation is applied per-lane in the async gather addresses.
// Output accumulates each variant / 3 at the LOGICAL row index (the
// reference does not un-permute col_attn).
// ---------------------------------------------------------------------------
__global__ __launch_bounds__(256, 1)
void attn_wmma(const float* __restrict__ qkv, float* __restrict__ out,
               int permuted, int dual) {
  __shared__ float kBuf[2][16 * 64];
  __shared__ float vBuf[2][16 * 64];
  __shared__ float pLds[AW][16 * 16];

  const int tid  = threadIdx.x;       // 0..255
  const int lane = tid & 31;
  const int wid  = tid >> 5;          // 0..7
  const int half = lane >> 4;
  const int mr   = lane & 15;
  const int qt = blockIdx.x * AW + wid;   // 0..63
  const int h  = blockIdx.y;
  const int b  = blockIdx.z;
  const float* base = qkv + (size_t)b * SEQ * QKV_LD + (size_t)h * (3 * HDIM);

  const unsigned int kB0 = (unsigned int)(uintptr_t)(&kBuf[0][0]);
  const unsigned int kB1 = (unsigned int)(uintptr_t)(&kBuf[1][0]);
  const unsigned int vB0 = (unsigned int)(uintptr_t)(&vBuf[0][0]);
  const unsigned int vB1 = (unsigned int)(uintptr_t)(&vBuf[1][0]);
  const int srow = tid >> 4;                    // panel row 0..15
  const int scol = (tid & 15) << 2;             // panel col (floats)
  const unsigned int slotOff = (unsigned int)(tid << 4);

  // ---- preload Q fragments (A layout, 16 k-steps of 4) ----
  const int qsrc = permrow(qt * 16 + mr, permuted);
  const float* qptr = base + (size_t)qsrc * QKV_LD + 2 * half;
  v2f aq[16];
#pragma unroll
  for (int d = 0; d < 16; ++d)
    aq[d] = *reinterpret_cast<const v2f*>(qptr + d * 4);

  const v8f zero = {0.f, 0.f, 0.f, 0.f, 0.f, 0.f, 0.f, 0.f};
  v8f acc0[4] = {zero, zero, zero, zero};
  v8f acc1[4] = {zero, zero, zero, zero};
  float m0v[8], l0v[8], m1v[8], l1v[8];
#pragma unroll
  for (int r = 0; r < 8; ++r) {
    m0v[r] = -INFINITY; l0v[r] = 0.0f;
    m1v[r] = -INFINITY; l1v[r] = 0.0f;
  }
  const int mask0 = dual ? 0 : 1;     // dual: set0=row-mask; single: blk-mask

  // Prime: K/V panel 0 -> buffer 0 (per-lane gather handles permutation)
  {
    const float* rowp = base + (size_t)permrow(srow, permuted) * QKV_LD;
    async_copy_b128(kB0 + slotOff, rowp + HDIM + scol);
    async_copy_b128(vB0 + slotOff, rowp + 2 * HDIM + scol);
  }

  for (int kt = 0; kt < SEQ / 16; ++kt) {
    const int cur = kt & 1;
    if (kt + 1 < SEQ / 16) {
      const float* rowp =
          base + (size_t)permrow((kt + 1) * 16 + srow, permuted) * QKV_LD;
      async_copy_b128((cur ? kB0 : kB1) + slotOff, rowp + HDIM + scol);
      async_copy_b128((cur ? vB0 : vB1) + slotOff, rowp + 2 * HDIM + scol);
      wait_async_le2();
    } else {
      wait_async_le0();
    }
    __syncthreads();                  // publish K/V panel across waves

    const float* kb = &kBuf[cur][0];
    const float* vb = &vBuf[cur][0];

    // ---- S tile = Q (16x64) @ K^T (64x16); K fragment = contiguous b64 ----
    v8f s = zero;
    const float* kfrag = kb + mr * 64 + 2 * half;
#pragma unroll
    for (int d = 0; d < 16; ++d) {
      v2f bk = *reinterpret_cast<const v2f*>(kfrag + d * 4);
      s = wmma_f32(aq[d], bk, s);
    }

    softmax_pv(s, mask0, qt, kt, mr, half, m0v, l0v, acc0, &pLds[wid][0], vb);
    if (dual)
      softmax_pv(s, 1, qt, kt, mr, half, m1v, l1v, acc1, &pLds[wid][0], vb);

    __syncthreads();                  // done reading this K/V buffer
  }

  // ---- epilogue: out[logical row, h*64 + col] += sum_of_variants / 3 ----
#pragma unroll
  for (int r = 0; r < 8; ++r) {
    const int i = qt * 16 + r + 8 * half;
    const float inv0 = 1.0f / (3.0f * l0v[r]);
    const float inv1 = dual ? (1.0f / (3.0f * l1v[r])) : 0.0f;
    float* op = out + ((size_t)(b * SEQ + i)) * DMODEL + h * HDIM + mr;
#pragma unroll
    for (int nt = 0; nt < 4; ++nt) {
      float v = acc0[nt][r] * inv0;
      if (dual) v += acc1[nt][r] * inv1;
      op[nt * 16] += v;
    }
  }
}

// ---------------------------------------------------------------------------
// Residual + LayerNorm (in place on x): x = LN(x + t) * g + b
// One block (128 threads) per row of 512; two-pass variance like reference.
// ---------------------------------------------------------------------------
__global__ void add_ln_kernel(float* __restrict__ x, const float* __restrict__ t,
                              const float* __restrict__ g, const float* __restrict__ bta) {
  __shared__ float sv[DMODEL];
  __shared__ float red[128];
  const int row = blockIdx.x;
  const int tid = threadIdx.x;
  float* xr = x + (size_t)row * DMODEL;
  const float* tr = t + (size_t)row * DMODEL;

  float local = 0.0f;
#pragma unroll
  for (int e = 0; e < 4; ++e) {
    int c = tid + e * 128;
    float s = xr[c] + tr[c];
    sv[c] = s;
    local += s;
  }
  red[tid] = local;
  __syncthreads();
  for (int st = 64; st > 0; st >>= 1) {
    if (tid < st) red[tid] += red[tid + st];
    __syncthreads();
  }
  const float mean = red[0] * (1.0f / DMODEL);
  __syncthreads();

  float l2 = 0.0f;
#pragma unroll
  for (int e = 0; e < 4; ++e) {
    int c = tid + e * 128;
    float d = sv[c] - mean;
    l2 += d * d;
  }
  red[tid] = l2;
  __syncthreads();
  for (int st = 64; st > 0; st >>= 1) {
    if (tid < st) red[tid] += red[tid + st];
    __syncthreads();
  }
  const float rstd = rsqrtf(red[0] * (1.0f / DMODEL) + 1e-5f);
  __syncthreads();

#pragma unroll
  for (int e = 0; e < 4; ++e) {
    int c = tid + e * 128;
    xr[c] = (sv[c] - mean) * rstd * g[c] + bta[c];
  }
}

// ---------------------------------------------------------------------------
// Host orchestration
// ---------------------------------------------------------------------------
extern "C" void kernel_launch(void* const* d_in, const int* in_sizes, int n_in,
                              void* d_out, int out_size, void* d_ws, size_t ws_size,
                              hipStream_t stream) {
  (void)in_sizes; (void)n_in; (void)out_size; (void)ws_size;

  const int*   tokens   = (const int*)  d_in[0];
  const float* embed_w  = (const float*)d_in[1];
  const float* pos_e    = (const float*)d_in[2];
  const float* qkv_w    = (const float*)d_in[3];   // [L,512,1536]
  const float* attn_ow  = (const float*)d_in[4];   // [L,512,512]
  const float* ff_w1    = (const float*)d_in[5];   // [L,512,2048]
  const float* ff_b1    = (const float*)d_in[6];   // [L,2048]
  const float* ff_w2    = (const float*)d_in[7];   // [L,2048,512]
  const float* ff_b2    = (const float*)d_in[8];   // [L,512]
  const float* ln1_g    = (const float*)d_in[9];
  const float* ln1_b    = (const float*)d_in[10];
  const float* ln2_g    = (const float*)d_in[11];
  const float* ln2_b    = (const float*)d_in[12];
  const float* out_w    = (const float*)d_in[13];  // [512,32000]
  float* out = (float*)d_out;

  // Workspace layout (bytes)
  char* ws = (char*)d_ws;
  float* x    = (float*)(ws);                          // 2048x512   =  4 MB
  float* qkv  = (float*)(ws + (4u  << 20));            // 2048x1536  = 12 MB
  float* aacc = (float*)(ws + (16u << 20));            // 2048x512   =  4 MB
  float* tmp  = (float*)(ws + (20u << 20));            // 2048x512   =  4 MB
  float* ff   = (float*)(ws + (24u << 20));            // 2048x2048  = 16 MB

  const dim3 gblk(256, 1, 1);                          // 8 waves per block

  // Embedding
  embed_kernel<<<dim3(ROWS), dim3(128), 0, stream>>>(tokens, embed_w, pos_e, x);

  for (int l = 0; l < NLAYERS; ++l) {
    const float* qw = qkv_w   + (size_t)l * DMODEL * QKV_LD;
    const float* ow = attn_ow + (size_t)l * DMODEL * DMODEL;
    const float* w1 = ff_w1   + (size_t)l * DMODEL * FFDIM;
    const float* b1 = ff_b1   + (size_t)l * FFDIM;
    const float* w2 = ff_w2   + (size_t)l * FFDIM * DMODEL;
    const float* b2 = ff_b2   + (size_t)l * DMODEL;
    const float* g1 = ln1_g + (size_t)l * DMODEL;
    const float* be1 = ln1_b + (size_t)l * DMODEL;
    const float* g2 = ln2_g + (size_t)l * DMODEL;
    const float* be2 = ln2_b + (size_t)l * DMODEL;

    // qkv = x @ qw   [2048,512]x[512,1536]
    gemm_wmma<<<dim3(QKV_LD / GN, ROWS / 256), gblk, 0, stream>>>(
        x, qw, qkv, nullptr, DMODEL, DMODEL, QKV_LD, QKV_LD, 0);

    // attention accumulator = 0
    fill_zero<<<dim3((ROWS * DMODEL + 255) / 256), dim3(256), 0, stream>>>(
        aacc, ROWS * DMODEL);

    // attn #1 (row mask) + #3 (blk mask) share scores -> dual pass;
    // attn #2 = blk mask on permuted rows.
    attn_wmma<<<dim3(SEQ / 16 / AW, NHEADS, BATCH), gblk, 0, stream>>>(
        qkv, aacc, /*permuted=*/0, /*dual=*/1);
    attn_wmma<<<dim3(SEQ / 16 / AW, NHEADS, BATCH), gblk, 0, stream>>>(
        qkv, aacc, /*permuted=*/1, /*dual=*/0);

    // tmp = a @ ow
    gemm_wmma<<<dim3(DMODEL / GN, ROWS / 256), gblk, 0, stream>>>(
        aacc, ow, tmp, nullptr, DMODEL, DMODEL, DMODEL, DMODEL, 0);

    // x = LN(x + tmp)
    add_ln_kernel<<<dim3(ROWS), dim3(128), 0, stream>>>(x, tmp, g1, be1);

    // ff = relu(x @ w1 + b1)
    gemm_wmma<<<dim3(FFDIM / GN, ROWS / 256), gblk, 0, stream>>>(
        x, w1, ff, b1, DMODEL, DMODEL, FFDIM, FFDIM, 1);

    // tmp = ff @ w2 + b2
    gemm_wmma<<<dim3(DMODEL / GN, ROWS / 256), gblk, 0, stream>>>(
        ff, w2, tmp, b2, FFDIM, FFDIM, DMODEL, DMODEL, 0);

    // x = LN(x + tmp)
    add_ln_kernel<<<dim3(ROWS), dim3(128), 0, stream>>>(x, tmp, g2, be2);
  }

  // out = x @ out_w   [2048,512]x[512,32000]  (out_w streams from L2)
  gemm_wmma<<<dim3(VOCAB / GN, ROWS / 256), gblk, 0, stream>>>(
      x, out_w, out, nullptr, DMODEL, DMODEL, VOCAB, VOCAB, 0);
}